// CovLayer_79422535238049
// MI455X (gfx1250) — compile-verified
//
#include <hip/hip_runtime.h>
#include <hip/hip_bf16.h>

typedef float v2f __attribute__((ext_vector_type(2)));
typedef float v8f __attribute__((ext_vector_type(8)));

#define C_IN   32
#define C_OUT  32
#define H_IN   512
#define W_IN   512
#define KSZ    288        // C_IN * 9
#define TH     4          // output rows per block
#define TW     64         // output cols per block
#define XCOLS  66         // TW + 2 halo
#define XROWS  6          // TH + 2 halo
#define XS_ROW 136        // 68 cols (padded) * 2 (c parity), in dwords
#define XS_CP  (XROWS * XS_ROW)   // dwords per channel-pair = 816

#if defined(__has_builtin)
#  if __has_builtin(__builtin_amdgcn_global_load_async_to_lds_b32)
#    define USE_ASYNC_LDS 1
#  endif
#endif

typedef __attribute__((address_space(1))) int* gint_t;   // global (AS1) int*
typedef __attribute__((address_space(3))) int* lint_t;   // LDS (AS3) int*

__device__ __forceinline__ void wait_async_zero() {
#if defined(__has_builtin)
#  if __has_builtin(__builtin_amdgcn_s_wait_asynccnt)
  __builtin_amdgcn_s_wait_asynccnt(0);
  return;
#  else
  asm volatile("s_wait_asynccnt 0x0" ::: "memory");
  return;
#  endif
#else
  asm volatile("s_wait_asynccnt 0x0" ::: "memory");
#endif
}

// Reorder weights from [och][c][tap] (OIHW flat) to [och][tap*32 + c] so the
// per-step A fragment (k, k+1 contiguous) is one aligned b64 load.
__global__ __launch_bounds__(256) void reorder_weights(
    const float* __restrict__ w, float* __restrict__ wre) {
  int t = blockIdx.x * 256 + threadIdx.x;
  if (t >= C_OUT * KSZ) return;
  int och = t / KSZ;
  int k   = t - och * KSZ;
  int tap = k >> 5;        // 0..8
  int c   = k & 31;        // 0..31
  wre[t] = w[och * KSZ + c * 9 + tap];
}

__global__ __launch_bounds__(256) void conv3x3_wmma_f32(
    const float* __restrict__ x,
    const float* __restrict__ wre,
    const float* __restrict__ bias,
    float* __restrict__ out) {
  // Input tile in LDS: channel-pair major, parity interleaved:
  // index = (c>>1)*XS_CP + row*XS_ROW + col*2 + (c&1)
  __shared__ float xs[(C_IN / 2) * XS_CP];   // 52224 B

  const int n  = blockIdx.z;
  const int h0 = blockIdx.y * TH;
  const int w0 = blockIdx.x * TW;
  const int tid = threadIdx.x;

  // ---- Stage input tile (zero-padded halo) ----
  // Direct memory -> LDS async copies (ASYNCcnt path); halo lanes store 0.
  for (int t = tid; t < C_IN * XROWS * XCOLS; t += 256) {
    int c   = t / (XROWS * XCOLS);
    int rem = t - c * (XROWS * XCOLS);
    int row = rem / XCOLS;
    int col = rem - row * XCOLS;
    int gh = h0 + row - 1;
    int gw = w0 + col - 1;
    float* ldst = &xs[(c >> 1) * XS_CP + row * XS_ROW + col * 2 + (c & 1)];
    if ((unsigned)gh < (unsigned)H_IN && (unsigned)gw < (unsigned)W_IN) {
      const float* gsrc = &x[(((size_t)n * C_IN + c) * H_IN + gh) * W_IN + gw];
#ifdef USE_ASYNC_LDS
      __builtin_amdgcn_global_load_async_to_lds_b32(
          (gint_t)gsrc, (lint_t)ldst, 0, 0);
#else
      *ldst = *gsrc;
#endif
    } else {
      *ldst = 0.0f;
    }
  }
#ifdef USE_ASYNC_LDS
  wait_async_zero();   // drain this wave's async copies before the barrier
#endif
  __syncthreads();

  const int wave = tid >> 5;
  const int lane = tid & 31;
  const int half = lane >> 4;   // lane group: selects K = 2*half + {0,1}
  const int lm   = lane & 15;   // A: M row / B,C,D: N column
  const int m    = wave & 1;    // which 16-channel half of och
  const int rt   = wave >> 1;   // output row within tile (0..3)

  v8f acc[4] = {};              // 4 N-subtiles of 16 cols each (64 cols)

  const int och_a = m * 16 + lm;                 // A-matrix row this lane feeds
  const float* __restrict__ wbase = wre + (size_t)och_a * KSZ + 2 * half;

  for (int tap = 0; tap < 9; ++tap) {
    const int tr = tap / 3;
    const int s  = tap - tr * 3;
    const int rowoff = (rt + tr) * XS_ROW;
#pragma unroll
    for (int cc = 0; cc < 32; cc += 4) {
      const int k0 = tap * 32 + cc;
      // A fragment: weight[och_a][k0 + 2*half + {0,1}]  (broadcast, cached)
      v2f a = *(const v2f*)(wbase + k0);
      // B base for this lane: channel pair (cc + 2*half)/2
      const float* bbase = &xs[((cc + 2 * half) >> 1) * XS_CP + rowoff];
#pragma unroll
      for (int sub = 0; sub < 4; ++sub) {
        const int col = sub * 16 + lm + s;       // 0..65
        v2f bf = *(const v2f*)(bbase + col * 2);
        acc[sub] = __builtin_amdgcn_wmma_f32_16x16x4_f32(
            false, a, false, bf, (short)0, acc[sub], false, false);
      }
    }
  }

  // ---- Epilogue: C/D layout -> NCHW + bias ----
  // VGPR g: lanes 0-15 hold M=g, lanes 16-31 hold M=8+g; N = lm.
#pragma unroll
  for (int g = 0; g < 8; ++g) {
    const int och = m * 16 + g + 8 * half;
    const float bv = bias[och];
    float* orow = out + (((size_t)n * C_OUT + och) * H_IN + (h0 + rt)) * W_IN
                      + w0 + lm;
#pragma unroll
    for (int sub = 0; sub < 4; ++sub) {
      orow[sub * 16] = acc[sub][g] + bv;
    }
  }
}

extern "C" void kernel_launch(void* const* d_in, const int* in_sizes, int n_in,
                              void* d_out, int out_size, void* d_ws, size_t ws_size,
                              hipStream_t stream) {
  const float* x    = (const float*)d_in[0];
  const float* wgt  = (const float*)d_in[1];
  const float* bias = (const float*)d_in[2];
  float* out = (float*)d_out;
  float* wre = (float*)d_ws;   // 32*288*4 = 36864 B of scratch

  // 1) Reorder weights into implicit-GEMM friendly layout.
  reorder_weights<<<(C_OUT * KSZ + 255) / 256, 256, 0, stream>>>(wgt, wre);

  // 2) Implicit-GEMM conv: grid = (W tiles, H tiles, batch).
  dim3 grid(W_IN / TW, H_IN / TH, 8);
  conv3x3_wmma_f32<<<grid, 256, 0, stream>>>(x, wre, bias, out);
}